// LSTMPointerNet_89361089560608
// MI455X (gfx1250) — compile-verified
//
#include <hip/hip_runtime.h>
#include <cmath>

typedef __attribute__((ext_vector_type(2))) float v2f;
typedef __attribute__((ext_vector_type(8))) float v8f;

__device__ __forceinline__ float sigf(float x) { return 1.0f / (1.0f + __expf(-x)); }

// ---------------------------------------------------------------------------
// Kernel 0: materialize X[t][b][d] = (t==0 ? init_i[d] : lstm_in[b][t-1][d])
//           and bias_sum = b_ih + b_hh
// ---------------------------------------------------------------------------
__global__ void prep_kernel(const float* __restrict__ lstm_in,
                            const float* __restrict__ init_i,
                            const float* __restrict__ b_ih,
                            const float* __restrict__ b_hh,
                            float* __restrict__ X,
                            float* __restrict__ bias_sum) {
    int idx = blockIdx.x * blockDim.x + threadIdx.x;
    if (idx < 2048) bias_sum[idx] = b_ih[idx] + b_hh[idx];
    const int total = 64 * 16 * 512;
    for (int i = idx; i < total; i += gridDim.x * blockDim.x) {
        int d = i & 511;
        int rem = i >> 9;
        int b = rem & 15;
        int t = rem >> 4;
        X[i] = (t == 0) ? init_i[d] : lstm_in[((size_t)b * 63 + (t - 1)) * 512 + d];
    }
}

// ---------------------------------------------------------------------------
// Generic fp32 WMMA GEMM: C[M,N] = A[M,K] @ (BT ? B[N,K]^T : B[K,N]) (+bias[n])
// Each wave owns FOUR 16x16 C tiles (64 consecutive columns): the A fragment
// is loaded once per k-step and reused by 4 independent WMMA chains (4x less
// A traffic, 4-deep WMMA ILP). 4 waves/block -> 256 columns per block.
// Uses V_WMMA_F32_16X16X4_F32 (native fp32 matrix op on CDNA5).
// ---------------------------------------------------------------------------
template <bool BT>
__global__ void __launch_bounds__(128) gemm_wmma(const float* __restrict__ A,
                                                 const float* __restrict__ B,
                                                 const float* __restrict__ bias,
                                                 float* __restrict__ C,
                                                 int M, int N, int K) {
    const int lane = threadIdx.x & 31;
    const int wave = threadIdx.x >> 5;
    const int half = lane >> 4;   // 0: K=0,1  1: K=2,3 (A frag); row-half for C
    const int r    = lane & 15;
    const int m0   = blockIdx.x * 16;
    const int n0   = blockIdx.y * 256 + wave * 64;   // 4 tiles: n0, n0+16, n0+32, n0+48

    v8f acc0 = {0.f,0.f,0.f,0.f,0.f,0.f,0.f,0.f};
    v8f acc1 = {0.f,0.f,0.f,0.f,0.f,0.f,0.f,0.f};
    v8f acc2 = {0.f,0.f,0.f,0.f,0.f,0.f,0.f,0.f};
    v8f acc3 = {0.f,0.f,0.f,0.f,0.f,0.f,0.f,0.f};

    const float* Arow = A + (size_t)(m0 + r) * K;
    // Row pointers for BT (B is [N,K]; we need rows n0+r, n0+16+r, ...)
    const float* Bt0 = B + (size_t)(n0 +  0 + r) * K;
    const float* Bt1 = B + (size_t)(n0 + 16 + r) * K;
    const float* Bt2 = B + (size_t)(n0 + 32 + r) * K;
    const float* Bt3 = B + (size_t)(n0 + 48 + r) * K;

    for (int k = 0; k < K; k += 4) {
        const int kk = k + 2 * half;
        v2f a = *(const v2f*)(Arow + kk);
        v2f b0, b1, b2, b3;
        if (BT) {
            b0 = *(const v2f*)(Bt0 + kk);
            b1 = *(const v2f*)(Bt1 + kk);
            b2 = *(const v2f*)(Bt2 + kk);
            b3 = *(const v2f*)(Bt3 + kk);
        } else {
            const float* Br0 = B + (size_t)kk * N;
            const float* Br1 = B + (size_t)(kk + 1) * N;
            b0.x = Br0[n0 +  0 + r]; b0.y = Br1[n0 +  0 + r];
            b1.x = Br0[n0 + 16 + r]; b1.y = Br1[n0 + 16 + r];
            b2.x = Br0[n0 + 32 + r]; b2.y = Br1[n0 + 32 + r];
            b3.x = Br0[n0 + 48 + r]; b3.y = Br1[n0 + 48 + r];
        }
        acc0 = __builtin_amdgcn_wmma_f32_16x16x4_f32(false, a, false, b0, (short)0, acc0, false, false);
        acc1 = __builtin_amdgcn_wmma_f32_16x16x4_f32(false, a, false, b1, (short)0, acc1, false, false);
        acc2 = __builtin_amdgcn_wmma_f32_16x16x4_f32(false, a, false, b2, (short)0, acc2, false, false);
        acc3 = __builtin_amdgcn_wmma_f32_16x16x4_f32(false, a, false, b3, (short)0, acc3, false, false);
    }

#pragma unroll
    for (int i = 0; i < 8; ++i) {
        const int row = m0 + i + 8 * half;
        float* Crow = C + (size_t)row * N + n0 + r;
        Crow[ 0] = acc0[i] + (bias ? bias[n0 +  0 + r] : 0.0f);
        Crow[16] = acc1[i] + (bias ? bias[n0 + 16 + r] : 0.0f);
        Crow[32] = acc2[i] + (bias ? bias[n0 + 32 + r] : 0.0f);
        Crow[48] = acc3[i] + (bias ? bias[n0 + 48 + r] : 0.0f);
    }
}

// ---------------------------------------------------------------------------
// LSTM recurrence: single workgroup (1024 threads = 32 waves), 64 steps.
// gates[16,2048] = xproj[t] + h @ w_hh^T ; wave w computes the 16-column tile
// w of each of the four gate regions (i,f,g,o), so every lane holds all four
// gates for its 8 (batch,col) cells in registers. h lives in LDS (WMMA A
// frags read from LDS), c lives in per-thread registers across steps.
// query output stored as [b][t][h].
// ---------------------------------------------------------------------------
__global__ void __launch_bounds__(1024, 1) lstm_kernel(const float* __restrict__ xproj,
                                                       const float* __restrict__ w_hh,
                                                       const float* __restrict__ init_h,
                                                       const float* __restrict__ init_c,
                                                       float* __restrict__ query) {
    __shared__ float hsh[16 * 512];   // h state, [batch][H], 32 KB

    const int tid  = threadIdx.x;
    const int lane = tid & 31;
    const int wave = tid >> 5;        // 0..31
    const int half = lane >> 4;
    const int r    = lane & 15;
    const int hcol = wave * 16 + r;   // column within H for this lane

    float creg[8];
#pragma unroll
    for (int i = 0; i < 8; ++i) creg[i] = init_c[hcol];

    if (tid < 512) {
        float hv = init_h[tid];
        for (int b = 0; b < 16; ++b) hsh[b * 512 + tid] = hv;
    }
    __syncthreads();

    // B row pointers for the four gate tiles handled by this wave
    const float* B0 = w_hh + (size_t)(16 * (0 * 32 + wave) + r) * 512;
    const float* B1 = w_hh + (size_t)(16 * (1 * 32 + wave) + r) * 512;
    const float* B2 = w_hh + (size_t)(16 * (2 * 32 + wave) + r) * 512;
    const float* B3 = w_hh + (size_t)(16 * (3 * 32 + wave) + r) * 512;

    for (int t = 0; t < 64; ++t) {
        v8f g0 = {0.f,0.f,0.f,0.f,0.f,0.f,0.f,0.f};
        v8f g1 = {0.f,0.f,0.f,0.f,0.f,0.f,0.f,0.f};
        v8f g2 = {0.f,0.f,0.f,0.f,0.f,0.f,0.f,0.f};
        v8f g3 = {0.f,0.f,0.f,0.f,0.f,0.f,0.f,0.f};

        for (int k = 0; k < 512; k += 4) {
            const int kk = k + 2 * half;
            v2f a;
            a.x = hsh[r * 512 + kk];
            a.y = hsh[r * 512 + kk + 1];
            v2f b0 = *(const v2f*)(B0 + kk);
            v2f b1 = *(const v2f*)(B1 + kk);
            v2f b2 = *(const v2f*)(B2 + kk);
            v2f b3 = *(const v2f*)(B3 + kk);
            g0 = __builtin_amdgcn_wmma_f32_16x16x4_f32(false, a, false, b0, (short)0, g0, false, false);
            g1 = __builtin_amdgcn_wmma_f32_16x16x4_f32(false, a, false, b1, (short)0, g1, false, false);
            g2 = __builtin_amdgcn_wmma_f32_16x16x4_f32(false, a, false, b2, (short)0, g2, false, false);
            g3 = __builtin_amdgcn_wmma_f32_16x16x4_f32(false, a, false, b3, (short)0, g3, false, false);
        }

        float hloc[8];
#pragma unroll
        for (int i = 0; i < 8; ++i) {
            const int row = i + 8 * half;                       // batch index
            const float* xp = xproj + (size_t)(t * 16 + row) * 2048;
            float gi = g0[i] + xp[hcol];
            float gf = g1[i] + xp[512 + hcol];
            float gg = g2[i] + xp[1024 + hcol];
            float go = g3[i] + xp[1536 + hcol];
            float cc = sigf(gf) * creg[i] + sigf(gi) * tanhf(gg);
            creg[i] = cc;
            hloc[i] = sigf(go) * tanhf(cc);
        }
        __syncthreads();   // all waves done reading h for step t
#pragma unroll
        for (int i = 0; i < 8; ++i) {
            const int row = i + 8 * half;
            hsh[row * 512 + hcol] = hloc[i];
            query[((size_t)row * 64 + t) * 512 + hcol] = hloc[i];
        }
        __syncthreads();
    }
}

// ---------------------------------------------------------------------------
// Score: out[b,t,m] = sum_h v[h] * tanh(attn_feat[b,m,h] + q[b,t,h])
// One block per (b,t); 8 waves, each wave does 32 m-rows with a 512-wide dot
// (16 h per lane) + wave32 shuffle reduction. Transcendental-bound (134M tanh).
// ---------------------------------------------------------------------------
__global__ void __launch_bounds__(256) score_kernel(const float* __restrict__ attn_feat,
                                                    const float* __restrict__ qbuf,
                                                    const float* __restrict__ attn_v,
                                                    float* __restrict__ out) {
    const int bt = blockIdx.x;   // b*64 + t
    const int b  = bt >> 6;

    __shared__ float qrow[512];
    __shared__ float vsh[512];
    for (int i = threadIdx.x; i < 512; i += 256) {
        qrow[i] = qbuf[(size_t)bt * 512 + i];
        vsh[i]  = attn_v[i];
    }
    __syncthreads();

    const int wave = threadIdx.x >> 5;
    const int lane = threadIdx.x & 31;

    for (int mi = 0; mi < 32; ++mi) {
        const int m = wave * 32 + mi;
        const float* feat = attn_feat + ((size_t)b * 256 + m) * 512;
        float acc = 0.f;
#pragma unroll 4
        for (int j = 0; j < 16; ++j) {
            const int h = lane + 32 * j;
            acc += vsh[h] * tanhf(feat[h] + qrow[h]);
        }
#pragma unroll
        for (int off = 16; off > 0; off >>= 1) acc += __shfl_xor(acc, off, 32);
        if (lane == 0) out[(size_t)bt * 256 + m] = acc;
    }
}

// ---------------------------------------------------------------------------
// Launch
// ---------------------------------------------------------------------------
extern "C" void kernel_launch(void* const* d_in, const int* in_sizes, int n_in,
                              void* d_out, int out_size, void* d_ws, size_t ws_size,
                              hipStream_t stream) {
    const float* knowledge_state = (const float*)d_in[0];   // [16,256,512]
    // d_in[1] knowledge_num: unused by reference output
    const float* lstm_in = (const float*)d_in[2];           // [16,63,512]
    const float* init_h  = (const float*)d_in[3];           // [1,512]
    const float* init_c  = (const float*)d_in[4];           // [1,512]
    const float* init_i  = (const float*)d_in[5];           // [512]
    const float* w_ih    = (const float*)d_in[6];           // [2048,512]
    const float* w_hh    = (const float*)d_in[7];           // [2048,512]
    const float* b_ih    = (const float*)d_in[8];           // [2048]
    const float* b_hh    = (const float*)d_in[9];           // [2048]
    const float* attn_wm = (const float*)d_in[10];          // [512,512]
    const float* attn_wq = (const float*)d_in[11];          // [512,512]
    const float* attn_v  = (const float*)d_in[12];          // [512]
    float* out = (float*)d_out;                             // [16,64,256]

    float* ws        = (float*)d_ws;
    float* attn_feat = ws;                                  // 16*256*512
    float* X         = attn_feat + 16 * 256 * 512;          // 64*16*512
    float* bias_sum  = X + 64 * 16 * 512;                   // 2048
    float* xproj     = bias_sum + 2048;                     // 64*16*2048
    float* query     = xproj + 64 * 16 * 2048;              // 16*64*512 ([b][t][h])
    float* qbuf      = query + 16 * 64 * 512;               // 16*64*512

    // 0) build X and bias_sum
    prep_kernel<<<2048, 256, 0, stream>>>(lstm_in, init_i, b_ih, b_hh, X, bias_sum);

    // 1) attn_feat = knowledge_state @ attn_wm        (M=4096, N=512, K=512)
    gemm_wmma<false><<<dim3(256, 2), 128, 0, stream>>>(knowledge_state, attn_wm,
                                                       nullptr, attn_feat, 4096, 512, 512);

    // 2) xproj = X @ w_ih^T + (b_ih + b_hh)           (M=1024, N=2048, K=512)
    gemm_wmma<true><<<dim3(64, 8), 128, 0, stream>>>(X, w_ih, bias_sum, xproj,
                                                     1024, 2048, 512);

    // 3) LSTM recurrence (single WGP, h in LDS, WMMA per step) -> query[b][t][h]
    lstm_kernel<<<1, 1024, 0, stream>>>(xproj, w_hh, init_h, init_c, query);

    // 4) qbuf = query @ attn_wq                        (M=1024, N=512, K=512)
    gemm_wmma<false><<<dim3(64, 2), 128, 0, stream>>>(query, attn_wq, nullptr,
                                                      qbuf, 1024, 512, 512);

    // 5) additive-attention scores
    score_kernel<<<1024, 256, 0, stream>>>(attn_feat, qbuf, attn_v, out);
}